// MPSWavefunction_42314017800198
// MI455X (gfx1250) — compile-verified
//
#include <hip/hip_runtime.h>

typedef __attribute__((ext_vector_type(2))) float v2f;
typedef __attribute__((ext_vector_type(4))) float v4f;
typedef __attribute__((ext_vector_type(8))) float v8f;

#define MPS_L   64
#define MPS_P   4
#define MPS_D   128
#define TILE_B  16
#define ROWPAD  132          // 128 + 4 -> conflict-free LDS banks for b64/b128 access
#define NTILES  8            // 128 / 16
#define KSTEPS  32           // 128 / 4

// ---------------------------------------------------------------------------
// Repack site_tensors [L][P][D][D] row-major into exact WMMA A-fragment lane
// order:  packed[site][s][ntile][kstep][lane] = { M[k0][n], M[k0+1][n] }
// with k0 = 4*kstep + 2*(lane>>4), n = ntile*16 + (lane&15).
// One thread per float2; 64*4*8*32*32 = 2,097,152 elements = 16 MB in d_ws.
// ---------------------------------------------------------------------------
__global__ __launch_bounds__(256) void mps_repack(const float* __restrict__ st,
                                                  v2f* __restrict__ packed) {
    unsigned idx  = blockIdx.x * 256u + threadIdx.x;
    unsigned lane = idx & 31u;
    unsigned ks   = (idx >> 5) & 31u;
    unsigned nt   = (idx >> 10) & 7u;
    unsigned s    = (idx >> 13) & 3u;
    unsigned site = idx >> 15;

    unsigned n = nt * 16u + (lane & 15u);
    unsigned k = ks * 4u + ((lane >> 4) << 1);

    const float* m = st + ((size_t)(site * 4u + s) << 14);   // 128*128 per matrix
    v2f val;
    val.x = m[(size_t)k * 128u + n];
    val.y = m[(size_t)(k + 1u) * 128u + n];
    packed[idx] = val;
}

// ---------------------------------------------------------------------------
// Main chain kernel. One workgroup (8 wave32) per 16-sample batch tile.
// Wave w owns output N-tile w. v (16 x 128 f32) lives in LDS, double-buffered.
// WMMA: A = transposed matrix tile (n x k), B = v^T (k x batch), C[n, batch].
// Compute against all 4 occupation matrices, branchless per-row select.
// ---------------------------------------------------------------------------
__global__ __launch_bounds__(256) void mps_chain(const int*   __restrict__ onstate,
                                                 const v2f*   __restrict__ packed,
                                                 const float* __restrict__ lvec,
                                                 const float* __restrict__ rvec,
                                                 float*       __restrict__ out) {
    __shared__ float vbuf[2][TILE_B * ROWPAD];
    __shared__ int   sbuf[MPS_L * TILE_B];
    __shared__ float red[TILE_B][16];

    const int tid  = threadIdx.x;
    const int lane = tid & 31;
    const int wave = tid >> 5;        // N-tile index 0..7
    const int n_lo = lane & 15;
    const int k_hi = lane >> 4;
    const int b0   = blockIdx.x * TILE_B;

    // v[row][col] = left_vec[col] broadcast, buffer 0
    for (int i = tid; i < TILE_B * MPS_D; i += 256) {
        int row = i >> 7, col = i & 127;
        vbuf[0][row * ROWPAD + col] = lvec[col];
    }
    // stage all 64 occupation indices for the 16 rows: sbuf[site][row]
    for (int i = tid; i < TILE_B * MPS_L; i += 256) {
        int row = i >> 6, site = i & 63;
        sbuf[site * TILE_B + row] = onstate[(b0 + row) * MPS_L + site];
    }
    __syncthreads();

    int buf = 0;
    for (int site = 0; site < MPS_L; ++site) {
        // A-fragment base for (site, s=0, ntile=wave), this lane
        const v2f* ap = packed + ((size_t)((site * 4) * NTILES + wave) << 10) + lane;
        if (site + 1 < MPS_L)
            __builtin_prefetch((const void*)(ap + ((size_t)(4 * NTILES) << 10)), 0, 0);

        // occupation for this lane's batch row; load early so the DS latency
        // hides under the WMMA stream
        const int s = sbuf[site * TILE_B + n_lo];

        v8f acc0 = {}, acc1 = {}, acc2 = {}, acc3 = {};
        const float* vrow = &vbuf[buf][n_lo * ROWPAD];

#pragma unroll 4
        for (int ks = 0; ks < KSTEPS; ++ks) {
            // v^T fragment: lane needs v[batch=n_lo][k = 4*ks + 2*k_hi + {0,1}]
            v2f bf = *(const v2f*)(vrow + (ks << 2) + (k_hi << 1));
            v2f a0 = ap[ks * 32];
            v2f a1 = ap[(1 * NTILES << 10) + ks * 32];
            v2f a2 = ap[(2 * NTILES << 10) + ks * 32];
            v2f a3 = ap[(3 * NTILES << 10) + ks * 32];
            acc0 = __builtin_amdgcn_wmma_f32_16x16x4_f32(false, a0, false, bf, (short)0, acc0, false, false);
            acc1 = __builtin_amdgcn_wmma_f32_16x16x4_f32(false, a1, false, bf, (short)0, acc1, false, false);
            acc2 = __builtin_amdgcn_wmma_f32_16x16x4_f32(false, a2, false, bf, (short)0, acc2, false, false);
            acc3 = __builtin_amdgcn_wmma_f32_16x16x4_f32(false, a3, false, bf, (short)0, acc3, false, false);
        }

        // branchless per-batch-row occupation select:
        // r = sum_s [s == s_b] * acc_s  (scalar cndmask weights + vector FMAs,
        // no EXEC-mask control flow)
        const float w0 = (s == 0) ? 1.0f : 0.0f;
        const float w1 = (s == 1) ? 1.0f : 0.0f;
        const float w2 = (s == 2) ? 1.0f : 0.0f;
        const float w3 = (s == 3) ? 1.0f : 0.0f;
        v8f r = acc0 * w0;
        r += acc1 * w1;
        r += acc2 * w2;
        r += acc3 * w3;

        // write new v tile: row = n_lo, cols = wave*16 + k_hi*8 + j
        float* dst = &vbuf[buf ^ 1][n_lo * ROWPAD + (wave << 4) + (k_hi << 3)];
        *(v4f*)(dst + 0) = __builtin_shufflevector(r, r, 0, 1, 2, 3);
        *(v4f*)(dst + 4) = __builtin_shufflevector(r, r, 4, 5, 6, 7);

        __syncthreads();
        buf ^= 1;
    }

    // amp[b] = v[b,:] . right_vec  (256 threads -> 16 rows x 16 lane-partials)
    {
        int row = wave * 2 + k_hi;
        const float* vr = &vbuf[buf][row * ROWPAD + n_lo * 8];
        const float* rv = rvec + n_lo * 8;
        float p = 0.f;
#pragma unroll
        for (int c = 0; c < 8; ++c) p += vr[c] * rv[c];
        red[row][n_lo] = p;
    }
    __syncthreads();
    if (tid < TILE_B) {
        float a = 0.f;
#pragma unroll
        for (int i = 0; i < 16; ++i) a += red[tid][i];
        out[b0 + tid] = a;
    }
}

// ---------------------------------------------------------------------------
extern "C" void kernel_launch(void* const* d_in, const int* in_sizes, int n_in,
                              void* d_out, int out_size, void* d_ws, size_t ws_size,
                              hipStream_t stream) {
    const int*   onstate = (const int*)d_in[0];     // [B, L] int32
    const float* st      = (const float*)d_in[1];   // [L, P, D, D] f32
    const float* lvec    = (const float*)d_in[2];   // [D]
    const float* rvec    = (const float*)d_in[3];   // [D]
    float*       out     = (float*)d_out;           // [B]
    v2f*         packed  = (v2f*)d_ws;              // 16 MB repacked A-fragments

    const int B = in_sizes[0] / MPS_L;              // 2048

    const int repack_elems = MPS_L * MPS_P * NTILES * KSTEPS * 32;  // 2,097,152
    mps_repack<<<repack_elems / 256, 256, 0, stream>>>(st, packed);
    mps_chain<<<B / TILE_B, 256, 0, stream>>>(onstate, packed, lvec, rvec, out);
}